// Antecedents_11647951306867
// MI455X (gfx1250) — compile-verified
//
#include <hip/hip_runtime.h>

typedef __attribute__((ext_vector_type(2))) float v2f;
typedef __attribute__((ext_vector_type(4))) float v4f;
typedef __attribute__((ext_vector_type(8))) float v8f;

#define N_SAMPLES 16384
#define N_RULES   4096
#define WAVES_PER_BLOCK 8

// One wave32 per sample. Rules split r = rHigh*64 + rLow:
//   a_high[rHigh] = m0[s0]*m1[s1]*m2[s2]   (rHigh = s0*16 + s1*4 + s2)
//   b_low [rLow ] = m3[s3]*m4[s4]*m5[s5]   (rLow  = s3*16 + s4*4 + s5)
// out[n, rHigh*64 + rLow] = a_high * b_low  -> 64x64 rank-1 outer product,
// computed as 16 WMMA f32 16x16x4 tiles with K zero-padded to 1 effective term.
// A (M dim) carries b_low so M is the fast output digit -> per-lane contiguous
// 8-float runs -> two aligned b128 non-temporal stores per tile.
__global__ __launch_bounds__(256, 1)
void fuzzy_antecedents_wmma(const float* __restrict__ mem, float* __restrict__ out) {
    const int lane = threadIdx.x & 31;
    const int n    = blockIdx.x * WAVES_PER_BLOCK + (threadIdx.x >> 5); // sample
    const int l16  = lane & 15;
    const bool lo  = lane < 16;

    const size_t VS = (size_t)N_SAMPLES * 4;       // stride between variables
    const float* base = mem + (size_t)n * 4;       // this sample's row

    // Variables 0 and 3 supply the tile-index digit: need all four sets.
    v4f m0 = *(const v4f*)(base + 0 * VS);
    v4f m3 = *(const v4f*)(base + 3 * VS);

    const int sMid  = (l16 >> 2) & 3;  // middle digit within a 16-wide tile
    const int sFast = l16 & 3;         // fastest digit within a 16-wide tile

    const float p12 = base[1 * VS + sMid] * base[2 * VS + sFast];
    const float p45 = base[4 * VS + sMid] * base[5 * VS + sFast];

    // a_high[jt*16 + l16] -> B matrix (N dimension)
    float ah0 = m0.x * p12, ah1 = m0.y * p12, ah2 = m0.z * p12, ah3 = m0.w * p12;
    // b_low[it*16 + l16]  -> A matrix (M dimension)
    float bl0 = m3.x * p45, bl1 = m3.y * p45, bl2 = m3.z * p45, bl3 = m3.w * p45;

    if (!lo) { // lanes 16-31 hold K=2 of A / K=2 of B: force zero so only K=0 acts
        ah0 = ah1 = ah2 = ah3 = 0.f;
        bl0 = bl1 = bl2 = bl3 = 0.f;
    }
    const float ah[4] = { ah0, ah1, ah2, ah3 };
    const float bl[4] = { bl0, bl1, bl2, bl3 };

    float* orow = out + (size_t)n * N_RULES;

    #pragma unroll
    for (int it = 0; it < 4; ++it) {               // rLow tile (A / M dim)
        v2f A; A.x = bl[it]; A.y = 0.f;            // VGPR1 (K=1,3) = 0
        #pragma unroll
        for (int jt = 0; jt < 4; ++jt) {           // rHigh tile (B / N dim)
            v2f B; B.x = ah[jt]; B.y = 0.f;
            v8f c = {};
            c = __builtin_amdgcn_wmma_f32_16x16x4_f32(
                    /*neg_a=*/false, A, /*neg_b=*/false, B,
                    /*c_mod=*/(short)0, c, /*reuse_a=*/false, /*reuse_b=*/false);

            // D(M,N) = bl[i+M] * ah[j+N]; r = (jt*16+N)*64 + it*16 + M
            const size_t r = (size_t)(jt * 16 + l16) * 64
                           + (size_t)(it * 16) + (lo ? 0u : 8u);
            v4f d0 = { c[0], c[1], c[2], c[3] };   // M = base+0..3
            v4f d1 = { c[4], c[5], c[6], c[7] };   // M = base+4..7
            __builtin_nontemporal_store(d0, (v4f*)(orow + r));
            __builtin_nontemporal_store(d1, (v4f*)(orow + r + 4));
        }
    }
}

extern "C" void kernel_launch(void* const* d_in, const int* in_sizes, int n_in,
                              void* d_out, int out_size, void* d_ws, size_t ws_size,
                              hipStream_t stream) {
    (void)in_sizes; (void)n_in; (void)out_size; (void)d_ws; (void)ws_size;
    const float* mem = (const float*)d_in[0];
    float* out = (float*)d_out;

    dim3 grid(N_SAMPLES / WAVES_PER_BLOCK);  // 2048 blocks
    dim3 block(256);                          // 8 waves of 32
    hipLaunchKernelGGL(fuzzy_antecedents_wmma, grid, block, 0, stream, mem, out);
}